// ObservationTokenizer_84232898609863
// MI455X (gfx1250) — compile-verified
//
#include <hip/hip_runtime.h>
#include <hip/hip_bf16.h>
#include <stddef.h>

// ---------------------------------------------------------------------------
// ObservationTokenizer for MI455X (gfx1250, wave32).
// Four tiny-K GEMMs (K = 8/12/16/32, N = 512) over gathered feature slices,
// fused bias add. f32 end-to-end via V_WMMA_F32_16X16X4_F32 (accumulating
// K-loop). Memory-bound (~856 MB of output stores) -> non-temporal stores.
// All geometry is template-constant so load/store offsets fold into the
// 24-bit instruction immediates (one address pair, clauses of NT stores).
// ---------------------------------------------------------------------------

typedef __attribute__((ext_vector_type(2))) float v2f;
typedef __attribute__((ext_vector_type(8))) float v8f;

#define OBS_DIM   643
#define D_MODEL   512
#define BATCH     8192

// One wave computes one 16x16 f32 output tile:
//   rows  = 16 consecutive batch indices (fixed token index, blockIdx.z)
//   cols  = one 16-wide slice of D_MODEL
// K-loop in steps of 4 using V_WMMA_F32_16X16X4_F32 with accumulation.
//
// A 16x4 f32 layout (ISA 7.12.2): lanes 0-15 -> {K=0 in v0, K=1 in v1},
// lanes 16-31 -> {K=2 in v0, K=3 in v1}; M = lane % 16.
// B 4x16 mirrored: N = lane % 16; lanes 0-15 hold K=0/1, lanes 16-31 K=2/3.
// C/D 16x16 f32: VGPR r -> M = r (lanes 0-15) / M = r + 8 (lanes 16-31),
// N = lane % 16.
//
// DFEAT    : K (feature dim), multiple of 4
// FEAT0    : column of first feature of token 0 in encoded_obs
// FSTRIDE  : columns between consecutive tokens of this type
// TOKSTRIDE: token rows per batch element in the output tensor
template <int DFEAT, int FEAT0, int FSTRIDE, int TOKSTRIDE>
__global__ __launch_bounds__(256) void tok_kernel(
    const float* __restrict__ enc,      // (BATCH, OBS_DIM)
    const float* __restrict__ W,        // (DFEAT, D_MODEL) row-major
    const float* __restrict__ bias,     // (D_MODEL)
    float* __restrict__ out_base)       // token (b, n) row at
                                        //   out_base + (b*TOKSTRIDE + n)*D_MODEL
{
    static_assert(DFEAT % 4 == 0, "K must be a multiple of 4");

    const int lane  = threadIdx.x & 31;
    const int wave  = threadIdx.x >> 5;
    const int jt    = (blockIdx.y << 3) + wave;   // N-tile index, 0..31
    const int b0    = blockIdx.x << 4;            // first batch row of M-tile
    const int tok   = blockIdx.z;                 // token index

    const int mrow  = lane & 15;                  // M (A) / N (B,C,D) sub-index
    const int khalf = (lane >> 4) << 1;           // 0 for lanes 0-15, 2 for 16-31
    const int ncol  = (jt << 4) + mrow;           // global output column

    // A rows: contiguous DFEAT floats per token; constant K offsets fold into
    // the load immediates (compiler merges the K/K+1 pair into b64 loads).
    const float* __restrict__ arow =
        enc + (size_t)(b0 + mrow) * OBS_DIM + (FEAT0 + tok * FSTRIDE) + khalf;
    const float* __restrict__ brow =
        W + (size_t)khalf * D_MODEL + ncol;

    v8f acc = {};
#pragma unroll
    for (int k0 = 0; k0 < DFEAT; k0 += 4) {
        v2f a;
        a.x = arow[k0];
        a.y = arow[k0 + 1];
        v2f bm;
        bm.x = brow[(size_t)k0 * D_MODEL];
        bm.y = brow[(size_t)(k0 + 1) * D_MODEL];
        // 8 args: (neg_a, A, neg_b, B, c_mod, C, reuse_a, reuse_b)
        acc = __builtin_amdgcn_wmma_f32_16x16x4_f32(
            false, a, false, bm, (short)0, acc, false, false);
    }

    const float bv = bias[ncol];

    // Single per-lane base address; the 8 row stores differ by the constant
    // TOKSTRIDE*D_MODEL*4 bytes (max 7*81920 B < 2^23), so they become one
    // clause of global_store_b32 with immediate offsets.
    const int mbase = (lane >> 4) << 3;           // 0 or 8
    float* __restrict__ p = out_base +
        ((size_t)(b0 + mbase) * TOKSTRIDE + tok) * D_MODEL + ncol;
#pragma unroll
    for (int r = 0; r < 8; ++r) {
        // Output (~856 MB) far exceeds L2 (192 MB): stream it non-temporally
        // so enc / W stay cache-resident.
        __builtin_nontemporal_store(acc[r] + bv, p + (size_t)r * (TOKSTRIDE * D_MODEL));
    }
}

extern "C" void kernel_launch(void* const* d_in, const int* in_sizes, int n_in,
                              void* d_out, int out_size, void* d_ws, size_t ws_size,
                              hipStream_t stream) {
    (void)in_sizes; (void)n_in; (void)out_size; (void)d_ws; (void)ws_size;

    const float* enc = (const float*)d_in[0];
    const float* Wb  = (const float*)d_in[1];
    const float* bb  = (const float*)d_in[2];
    const float* We  = (const float*)d_in[3];
    const float* be  = (const float*)d_in[4];
    const float* Ws  = (const float*)d_in[5];
    const float* bs  = (const float*)d_in[6];
    const float* Wn  = (const float*)d_in[7];
    const float* bn  = (const float*)d_in[8];
    float* out = (float*)d_out;

    // Output layout (flat, f32):
    //   ca_tokens : (B, 40, 512)  [batt tokens 0..19, ev tokens 20..39]
    //   sro_tok   : (B, 10, 512)
    //   nfc_tok   : (B,  1, 512)
    float* out_ca  = out;
    float* out_ev  = out + (size_t)20 * D_MODEL;                // token offset 20
    float* out_sro = out + (size_t)BATCH * 40 * D_MODEL;
    float* out_nfc = out + (size_t)BATCH * 50 * D_MODEL;

    const dim3 block(256);                        // 8 wave32 waves
    const unsigned gx = BATCH / 16;               // 512 M-tiles
    const unsigned gy = D_MODEL / (16 * 8);       // 4 (8 N-tiles per block)

    // batt: tokens at col n*9,   feats at n*9 + 1,  K=8
    tok_kernel<8,  1,   9,  40><<<dim3(gx, gy, 20), block, 0, stream>>>(enc, Wb, bb, out_ca);
    // ev:   tokens at 180+n*13,  feats at 181+n*13, K=12
    tok_kernel<12, 181, 13, 40><<<dim3(gx, gy, 20), block, 0, stream>>>(enc, We, be, out_ev);
    // sro:  tokens at 440+n*17,  feats at 441+n*17, K=16
    tok_kernel<16, 441, 17, 10><<<dim3(gx, gy, 10), block, 0, stream>>>(enc, Ws, bs, out_sro);
    // nfc:  single token, feats at 611..642, K=32
    tok_kernel<32, 611, 1,  1 ><<<dim3(gx, gy, 1 ), block, 0, stream>>>(enc, Wn, bn, out_nfc);
}